// ProteinGAT_89489938580249
// MI455X (gfx1250) — compile-verified
//
#include <hip/hip_runtime.h>
#include <hip/hip_bf16.h>
#include <math.h>

// ---------------------------------------------------------------------------
// ProteinGAT forward for MI455X (gfx1250, wave32, WMMA).
// GEMMs use V_WMMA_F32_16X16X4_F32 (full f32 precision, 16x16 C tiles).
// Edge softmax + aggregation via float atomics (bandwidth bound: ~0.7 GB).
// ---------------------------------------------------------------------------

#define NN 50000
#define EE 1200000
#define GG 25
#define FIN 128
#define HH 64

typedef __attribute__((ext_vector_type(2))) float v2f;
typedef __attribute__((ext_vector_type(8))) float v8f;

// ---------------- float atomic max (segment_max emulation) -----------------
__device__ __forceinline__ void atomicMaxF(float* addr, float v) {
    if (v >= 0.0f) {
        atomicMax((int*)addr, __float_as_int(v));
    } else {
        atomicMin((unsigned int*)addr, __float_as_uint(v));
    }
}

// ---------------- WMMA f32 GEMM: C[M x 64] = A[M x K] @ B[K x 64] (+bias) --
// One wave computes a 16x64 tile: 4 accumulators of 16x16 (v8f each).
// A fragment (16x4 f32): lanes 0-15 -> M=lane, K={k0,k0+1}; lanes 16-31 -> K={k0+2,k0+3}
// B fragment (4x16 f32): lanes 0-15 -> N=lane, K={k0,k0+1}; lanes 16-31 -> K={k0+2,k0+3}
// C tile: VGPR v holds (M = v + 8*half, N = lane&15).
__global__ void gemm_wmma_f32(const float* __restrict__ A,
                              const float* __restrict__ B,
                              const float* __restrict__ bias,
                              float* __restrict__ C,
                              int Mtiles, int K) {
    const int wave = threadIdx.x >> 5;
    const int lane = threadIdx.x & 31;
    const int tile = blockIdx.x * (blockDim.x >> 5) + wave;
    if (tile >= Mtiles) return;           // wave-uniform exit (EXEC all-1 below)

    const int row0 = tile * 16;
    const int half = lane >> 4;           // 0 or 1
    const int l16  = lane & 15;

    v8f acc0 = {}, acc1 = {}, acc2 = {}, acc3 = {};

    const int arow = row0 + l16;
    const float* __restrict__ Arow = A + (size_t)arow * K;

    for (int k0 = 0; k0 < K; k0 += 4) {
        const int kk = k0 + half * 2;
        v2f a;
        a[0] = Arow[kk];
        a[1] = Arow[kk + 1];

        const float* __restrict__ B0 = B + (size_t)kk * HH;
        const float* __restrict__ B1 = B + (size_t)(kk + 1) * HH;

        v2f b0; b0[0] = B0[l16];      b0[1] = B1[l16];
        v2f b1; b1[0] = B0[16 + l16]; b1[1] = B1[16 + l16];
        v2f b2; b2[0] = B0[32 + l16]; b2[1] = B1[32 + l16];
        v2f b3; b3[0] = B0[48 + l16]; b3[1] = B1[48 + l16];

        acc0 = __builtin_amdgcn_wmma_f32_16x16x4_f32(false, a, false, b0, (short)0, acc0, false, false);
        acc1 = __builtin_amdgcn_wmma_f32_16x16x4_f32(false, a, false, b1, (short)0, acc1, false, false);
        acc2 = __builtin_amdgcn_wmma_f32_16x16x4_f32(false, a, false, b2, (short)0, acc2, false, false);
        acc3 = __builtin_amdgcn_wmma_f32_16x16x4_f32(false, a, false, b3, (short)0, acc3, false, false);
    }

    // store: row = row0 + v + 8*half, col = nt*16 + l16
    #pragma unroll
    for (int v = 0; v < 8; ++v) {
        const int r = row0 + v + half * 8;
        float* __restrict__ Crow = C + (size_t)r * HH;
        float bv0 = bias ? bias[l16]      : 0.0f;
        float bv1 = bias ? bias[16 + l16] : 0.0f;
        float bv2 = bias ? bias[32 + l16] : 0.0f;
        float bv3 = bias ? bias[48 + l16] : 0.0f;
        Crow[l16]      = acc0[v] + bv0;
        Crow[16 + l16] = acc1[v] + bv1;
        Crow[32 + l16] = acc2[v] + bv2;
        Crow[48 + l16] = acc3[v] + bv3;
    }
}

// ---------------- c_edge[l] = dot(lin_edge_W[l], att_edge[l]) --------------
__global__ void cedge_kernel(const float* __restrict__ lin_edge_W,
                             const float* __restrict__ att_edge,
                             float* __restrict__ cedge) {
    const int l = blockIdx.x;     // 2 blocks
    const int t = threadIdx.x;    // 64 threads
    __shared__ float s[64];
    s[t] = lin_edge_W[l * HH + t] * att_edge[l * HH + t];
    __syncthreads();
    if (t < 32) s[t] += s[t + 32];
    __syncthreads();
    if (t == 0) {
        float sum = 0.0f;
        for (int i = 0; i < 32; ++i) sum += s[i];
        cedge[l] = sum;
    }
}

// ---------------- g = global_features @ W_glob + b_glob --------------------
__global__ void gvec_kernel(const float* __restrict__ GF,
                            const float* __restrict__ Wg,
                            const float* __restrict__ bg,
                            float* __restrict__ gvec) {
    const int idx = blockIdx.x * blockDim.x + threadIdx.x;
    if (idx >= GG * HH) return;
    const int gi = idx / HH, k = idx % HH;
    float acc = bg[k];
    #pragma unroll
    for (int j = 0; j < 6; ++j) acc += GF[gi * 6 + j] * Wg[j * HH + k];
    gvec[idx] = acc;
}

// ---------------- per-layer scratch init -----------------------------------
__global__ void layer_init_kernel(float* __restrict__ mbuf,
                                  float* __restrict__ dbuf,
                                  float* __restrict__ hnew) {
    const int idx = blockIdx.x * blockDim.x + threadIdx.x;
    if (idx < NN * HH) hnew[idx] = 0.0f;
    if (idx < NN) { mbuf[idx] = -INFINITY; dbuf[idx] = 0.0f; }
}

// ---------------- a_src / a_dst: one wave per node -------------------------
__global__ void att_kernel(const float* __restrict__ hs,
                           const float* __restrict__ att_src,
                           const float* __restrict__ att_dst,
                           float* __restrict__ a_src,
                           float* __restrict__ a_dst) {
    const int wid  = (blockIdx.x * blockDim.x + threadIdx.x) >> 5;
    const int lane = threadIdx.x & 31;
    if (wid >= NN) return;
    const float h0 = hs[(size_t)wid * HH + lane];
    const float h1 = hs[(size_t)wid * HH + lane + 32];
    float ps = h0 * att_src[lane] + h1 * att_src[lane + 32];
    float pd = h0 * att_dst[lane] + h1 * att_dst[lane + 32];
    #pragma unroll
    for (int off = 16; off > 0; off >>= 1) {
        ps += __shfl_xor(ps, off, 32);
        pd += __shfl_xor(pd, off, 32);
    }
    if (lane == 0) { a_src[wid] = ps; a_dst[wid] = pd; }
}

// ---------------- edge pass 1: leaky_relu score + segment max --------------
__global__ void edge_pass1(const long long* __restrict__ src,
                           const long long* __restrict__ dst,
                           const float* __restrict__ edge_attr,
                           const float* __restrict__ a_src,
                           const float* __restrict__ a_dst,
                           const float* __restrict__ cedge, int l,
                           float* __restrict__ alpha_raw,
                           float* __restrict__ mbuf) {
    const int e = blockIdx.x * blockDim.x + threadIdx.x;
    if (e >= EE) return;
    const long long s = src[e], d = dst[e];
    float v = a_src[s] + a_dst[d] + cedge[l] * edge_attr[e];
    v = (v > 0.0f) ? v : 0.2f * v;
    alpha_raw[e] = v;
    atomicMaxF(&mbuf[d], v);
}

// ---------------- edge pass 2: exp + segment sum ---------------------------
__global__ void edge_pass2(const long long* __restrict__ dst,
                           const float* __restrict__ alpha_raw,
                           const float* __restrict__ mbuf,
                           float* __restrict__ exbuf,
                           float* __restrict__ dbuf) {
    const int e = blockIdx.x * blockDim.x + threadIdx.x;
    if (e >= EE) return;
    const long long d = dst[e];
    const float ex = __expf(alpha_raw[e] - mbuf[d]);
    exbuf[e] = ex;
    atomicAdd(&dbuf[d], ex);
}

// ---------------- edge pass 3: message scatter (wave per edge) -------------
__global__ void message_kernel(const long long* __restrict__ src,
                               const long long* __restrict__ dst,
                               const float* __restrict__ exbuf,
                               const float* __restrict__ dbuf,
                               const float* __restrict__ hs,
                               float* __restrict__ hnew) {
    const int e    = (blockIdx.x * blockDim.x + threadIdx.x) >> 5;
    const int lane = threadIdx.x & 31;
    if (e >= EE) return;
    const long long s = src[e], d = dst[e];
    const float a = exbuf[e] / (dbuf[d] + 1e-16f);
    const float2 hv = ((const float2*)(hs + (size_t)s * HH))[lane];
    atomicAdd(&hnew[(size_t)d * HH + 2 * lane],     a * hv.x);
    atomicAdd(&hnew[(size_t)d * HH + 2 * lane + 1], a * hv.y);
}

// ---------------- finalize: h = relu(hnew + gat_bias[l]) -------------------
__global__ void node_finalize_kernel(const float* __restrict__ hnew,
                                     const float* __restrict__ gat_bias,
                                     float* __restrict__ h) {
    const int idx = blockIdx.x * blockDim.x + threadIdx.x;
    if (idx >= NN * HH) return;
    const int k = idx & (HH - 1);
    h[idx] = fmaxf(hnew[idx] + gat_bias[k], 0.0f);
}

// ---------------- pooling ---------------------------------------------------
__global__ void pool_zero_kernel(float* __restrict__ graph, float* __restrict__ cnt) {
    const int idx = blockIdx.x * blockDim.x + threadIdx.x;
    if (idx < GG * HH) graph[idx] = 0.0f;
    if (idx < GG) cnt[idx] = 0.0f;
}

__global__ void pool_kernel(const float* __restrict__ h,
                            const long long* __restrict__ batch,
                            float* __restrict__ graph,
                            float* __restrict__ cnt) {
    const int idx = blockIdx.x * blockDim.x + threadIdx.x;
    if (idx >= NN * HH) return;
    const int i = idx / HH, k = idx % HH;
    const long long b = batch[i];
    atomicAdd(&graph[b * HH + k], h[idx]);
    if (k == 0) atomicAdd(&cnt[b], 1.0f);
}

// ---------------- head MLP: out = relu([graph|g] @ W_comb + b) @ W_out + b -
__global__ void head_kernel(const float* __restrict__ graph,
                            const float* __restrict__ cnt,
                            const float* __restrict__ gvec,
                            const float* __restrict__ W_comb,
                            const float* __restrict__ b_comb,
                            const float* __restrict__ W_out,
                            const float* __restrict__ b_out,
                            float* __restrict__ out) {
    const int gi = blockIdx.x;    // 25 blocks
    const int t  = threadIdx.x;   // 64 threads
    __shared__ float cat[2 * HH];
    __shared__ float red[HH];
    cat[t]      = graph[gi * HH + t] / cnt[gi];
    cat[HH + t] = gvec[gi * HH + t];
    __syncthreads();
    float acc = b_comb[t];
    #pragma unroll 4
    for (int k = 0; k < 2 * HH; ++k) acc += cat[k] * W_comb[k * HH + t];
    acc = fmaxf(acc, 0.0f);
    red[t] = acc * W_out[t];
    __syncthreads();
    if (t < 32) red[t] += red[t + 32];
    __syncthreads();
    if (t == 0) {
        float s = 0.0f;
        for (int i = 0; i < 32; ++i) s += red[i];
        out[gi] = s + b_out[0];
    }
}

// ---------------------------------------------------------------------------
extern "C" void kernel_launch(void* const* d_in, const int* in_sizes, int n_in,
                              void* d_out, int out_size, void* d_ws, size_t ws_size,
                              hipStream_t stream) {
    (void)in_sizes; (void)n_in; (void)out_size; (void)ws_size;

    const float*     x          = (const float*)d_in[0];
    const float*     edge_attr  = (const float*)d_in[1];
    const float*     gfeat      = (const float*)d_in[2];
    const long long* edge_index = (const long long*)d_in[3];
    const long long* batch      = (const long long*)d_in[4];
    const float*     W_embed    = (const float*)d_in[5];
    const float*     b_embed    = (const float*)d_in[6];
    const float*     W_glob     = (const float*)d_in[7];
    const float*     b_glob     = (const float*)d_in[8];
    const float*     lin_W      = (const float*)d_in[9];
    const float*     att_src    = (const float*)d_in[10];
    const float*     att_dst    = (const float*)d_in[11];
    const float*     lin_edge_W = (const float*)d_in[12];
    const float*     att_edge   = (const float*)d_in[13];
    const float*     gat_bias   = (const float*)d_in[14];
    const float*     W_comb     = (const float*)d_in[15];
    const float*     b_comb     = (const float*)d_in[16];
    const float*     W_out      = (const float*)d_in[17];
    const float*     b_out      = (const float*)d_in[18];
    float*           out        = (float*)d_out;

    const long long* src = edge_index;
    const long long* dst = edge_index + EE;

    // workspace partition (floats)
    float* w      = (float*)d_ws;
    float* h      = w;                         // N*64
    float* hs     = h    + (size_t)NN * HH;    // N*64
    float* hnew   = hs   + (size_t)NN * HH;    // N*64
    float* a_src  = hnew + (size_t)NN * HH;    // N
    float* a_dst  = a_src + NN;                // N
    float* mbuf   = a_dst + NN;                // N
    float* dbuf   = mbuf  + NN;                // N
    float* exbuf  = dbuf  + NN;                // E
    float* gvec   = exbuf + EE;                // G*64
    float* graph  = gvec  + GG * HH;           // G*64
    float* cnt    = graph + GG * HH;           // G
    float* cedge  = cnt   + GG;                // 2

    const int Mtiles = NN / 16;                // 3125 exactly
    const int gemmBlocks = (Mtiles + 7) / 8;   // 8 waves / block

    // scalars + embeds
    cedge_kernel<<<2, 64, 0, stream>>>(lin_edge_W, att_edge, cedge);
    gemm_wmma_f32<<<gemmBlocks, 256, 0, stream>>>(x, W_embed, b_embed, h, Mtiles, FIN);
    gvec_kernel<<<(GG * HH + 255) / 256, 256, 0, stream>>>(gfeat, W_glob, b_glob, gvec);

    const int nThreads  = NN * HH;
    const int nBlocks   = (nThreads + 255) / 256;
    const int eBlocks   = (EE + 255) / 256;
    const int waveNodeB = (NN * 32 + 255) / 256;
    const int waveEdgeB = (EE * 32 + 255) / 256;

    for (int l = 0; l < 2; ++l) {
        layer_init_kernel<<<nBlocks, 256, 0, stream>>>(mbuf, dbuf, hnew);
        gemm_wmma_f32<<<gemmBlocks, 256, 0, stream>>>(h, lin_W + (size_t)l * HH * HH,
                                                      nullptr, hs, Mtiles, HH);
        att_kernel<<<waveNodeB, 256, 0, stream>>>(hs, att_src + l * HH, att_dst + l * HH,
                                                  a_src, a_dst);
        edge_pass1<<<eBlocks, 256, 0, stream>>>(src, dst, edge_attr, a_src, a_dst,
                                                cedge, l, exbuf /*alpha_raw reuse*/, mbuf);
        edge_pass2<<<eBlocks, 256, 0, stream>>>(dst, exbuf, mbuf, exbuf, dbuf);
        message_kernel<<<waveEdgeB, 256, 0, stream>>>(src, dst, exbuf, dbuf, hs, hnew);
        node_finalize_kernel<<<nBlocks, 256, 0, stream>>>(hnew, gat_bias + l * HH, h);
    }

    pool_zero_kernel<<<(GG * HH + 255) / 256, 256, 0, stream>>>(graph, cnt);
    pool_kernel<<<nBlocks, 256, 0, stream>>>(h, batch, graph, cnt);
    head_kernel<<<GG, 64, 0, stream>>>(graph, cnt, gvec, W_comb, b_comb, W_out, b_out, out);
}